// MGCN_TPN_Classifier_55310588838069
// MI455X (gfx1250) — compile-verified
//
#include <hip/hip_runtime.h>
#include <stdint.h>

// -------- problem constants (match reference) --------
constexpr int kNImg = 16;
constexpr int kSize = 512;
constexpr int kNSp  = 512;
constexpr int kWay  = 5;
constexpr int kTab  = kNSp * kWay;            // 2560 floats per image (10 KB)
constexpr int kPix  = kSize * kSize;          // 262144 pixels per image
constexpr int kBlocksPerImage = 128;
constexpr int kPixPerBlock = kPix / kBlocksPerImage;  // 2048
constexpr int kThreads = 256;                 // 8 wave32s

typedef uint32_t u32x4 __attribute__((ext_vector_type(4)));
typedef int      i32x4 __attribute__((ext_vector_type(4)));
typedef int      i32x8 __attribute__((ext_vector_type(8)));

__global__ __launch_bounds__(kThreads)
void sp_gather_kernel(const float* __restrict__ logits,
                      const int*   __restrict__ segments,
                      float*       __restrict__ out)
{
    __shared__ float tab[kTab];

    const int img   = blockIdx.y;
    const int chunk = blockIdx.x;

    const float* src = logits + (size_t)img * kTab;

#if __has_builtin(__builtin_amdgcn_tensor_load_to_lds)
    // ---- TDM: DMA this image's 512x5 f32 logits table into LDS ----
    // TDM ignores EXEC and issues once per wave; gate to wave 0 only
    // (threadIdx.x < 32 is wave-uniform in wave32, other waves branch away).
    if (threadIdx.x < 32u) {
        const uint64_t ga = (uint64_t)(uintptr_t)src;
        // Flat shared pointers are {LDS aperture, byte offset} (ISA 10.2):
        // low 32 bits are the wave-relative LDS byte address the D# wants.
        const uint32_t lba = (uint32_t)(uintptr_t)(const void*)&tab[0];

        // D# group 0 (128b): count=1 | is_restore=0 | gather=0 ;
        // lds_addr ; global_addr[56:0] ; type=2 ("image")
        u32x4 g0;
        g0[0] = 1u;
        g0[1] = lba;
        g0[2] = (uint32_t)ga;                               // global_addr[31:0]
        g0[3] = (uint32_t)((ga >> 32) & 0x1FFFFFFu)         // global_addr[56:32]
              | (2u << 30);                                 // type = 2
        // D# group 1 (256b): wg_mask=0 (not in cluster), data_size=2 (4B),
        // no atomic-barrier / iterate / pad; tensor_dim0 = tile_dim0 = 2560,
        // tensor_dim1 = tile_dim1 = 1 (single contiguous row).
        i32x8 g1;
        g1[0] = (int)(2u << 16);                            // data_size = 4B
        g1[1] = (int)((uint32_t)kTab << 16);                // tensor_dim0 lo16 @ [63:48]
        g1[2] = (int)(1u << 16);                            // tensor_dim1 = 1 @ [95:80]
        g1[3] = (int)((uint32_t)kTab << 16);                // tile_dim0 @ [127:112]
        g1[4] = 1;                                          // tile_dim1 = 1, tile_dim2 = 0
        g1[5] = kTab;                                       // tensor_dim0_stride lo32
        g1[6] = 0;
        g1[7] = 0;
        const i32x4 gz = {0, 0, 0, 0};
# if __has_include(<hip/amd_detail/amd_gfx1250_TDM.h>)
        // amdgpu-toolchain (clang-23): 6-arg form
        const i32x8 gz8 = {0, 0, 0, 0, 0, 0, 0, 0};
        __builtin_amdgcn_tensor_load_to_lds(g0, g1, gz, gz, gz8, 0);
# else
        // ROCm 7.2 (clang-22): 5-arg form
        __builtin_amdgcn_tensor_load_to_lds(g0, g1, gz, gz, 0);
# endif
# if __has_builtin(__builtin_amdgcn_s_wait_tensorcnt)
        __builtin_amdgcn_s_wait_tensorcnt(0);
# else
        asm volatile("s_wait_tensorcnt 0x0" ::: "memory");
# endif
    }
#else
    // Fallback: cooperative staged copy (still LDS-resident table).
    for (int i = threadIdx.x; i < kTab; i += kThreads) tab[i] = src[i];
#endif
    __syncthreads();

    // ---- gather: 1 pixel per thread per iteration ----
    const size_t pixBase = (size_t)img * kPix + (size_t)chunk * kPixPerBlock;
    const int*  seg = segments + pixBase;
    float*      dst = out + pixBase * kWay;

    for (int p = (int)threadIdx.x; p < kPixPerBlock; p += kThreads) {
        // Prefetch next tile of the index stream (speculative, drop-safe OOB).
        __builtin_prefetch(seg + p + kThreads, 0, 1);
        const int s = seg[p];                // coalesced b32 load
        // Unpadded stride-5 rows: bank = (5s+c) mod 64, gcd(5,64)=1 ->
        // near-uniform bank spread for random superpixel ids.
        const float* row = &tab[s * kWay];
        const float v0 = row[0];
        const float v1 = row[1];
        const float v2 = row[2];
        const float v3 = row[3];
        const float v4 = row[4];
        float* o = dst + (size_t)p * kWay;   // wave writes 640 contiguous bytes
        o[0] = v0; o[1] = v1; o[2] = v2; o[3] = v3; o[4] = v4;
    }
}

extern "C" void kernel_launch(void* const* d_in, const int* in_sizes, int n_in,
                              void* d_out, int out_size, void* d_ws, size_t ws_size,
                              hipStream_t stream) {
    (void)in_sizes; (void)n_in; (void)out_size; (void)d_ws; (void)ws_size;
    const float* logits   = (const float*)d_in[0];
    const int*   segments = (const int*)d_in[1];
    float*       out      = (float*)d_out;
    dim3 grid(kBlocksPerImage, kNImg);
    sp_gather_kernel<<<grid, kThreads, 0, stream>>>(logits, segments, out);
}